// HybridSymmetricLoss_53987738911515
// MI455X (gfx1250) — compile-verified
//
#include <hip/hip_runtime.h>
#include <math.h>

typedef __attribute__((ext_vector_type(2))) float v2f;
typedef __attribute__((ext_vector_type(8))) float v8f;

#define NB      65536
#define ROWLEN  100          // J*J
#define WAVES_PER_BLOCK 8
#define BATCHES_PER_WAVE 5
#define LN2     0.69314718055994530942f

__global__ void hsl_zero_kernel(float* out) { out[0] = 0.0f; }

// clipped natural log via bare v_log_f32 (hardware log2) * ln2.
// x in [0,1]: x==0 -> -inf -> clipped to -100 (matches jnp.clip(log(p),-100)).
__device__ __forceinline__ float clog(float x) {
    return fmaxf(__builtin_amdgcn_logf(x) * LN2, -100.0f);
}

__global__ __launch_bounds__(256) void hsl_main_kernel(
    const float* __restrict__ assign,      // [B,3,100] probabilities
    const float* __restrict__ category,    // [B,3]
    const float* __restrict__ labels,      // [B,3,100]
    const float* __restrict__ cat_labels,  // [B,3]
    float* __restrict__ out)
{
    __shared__ float ldsC[WAVES_PER_BLOCK][256];  // 16x16 C tile per wave
    __shared__ float ldsS[WAVES_PER_BLOCK][32];   // per-lane sum(l1) partials
    __shared__ float blockSum;

    const int tid  = threadIdx.x;
    const int wv   = tid >> 5;
    const int lane = tid & 31;
    if (tid == 0) blockSum = 0.0f;

    const int gwave  = blockIdx.x * WAVES_PER_BLOCK + wv;
    const int batch0 = gwave * BATCHES_PER_WAVE;

    // --- lane -> (batch, row, K-half) mapping for the block-diagonal pack ---
    const int q    = lane & 15;     // tile row/col index (0..15)
    const int h    = lane >> 4;     // K-half: h=0 -> K=0,1 ; h=1 -> K=2,3
    const int bsub = q / 3;         // sub-batch 0..4 (q==15 -> 5, redirected)
    const int r    = q - 3 * bsub;  // t for A rows, i for B cols
    const int b    = batch0 + bsub;
    const bool valid = (q < 15) && (b < NB);
    // Invalid lanes read row 0: their (finite) garbage only reaches
    // off-block-diagonal C entries / unread ldsS slots, which the finalizer
    // never touches -- so no masking is needed anywhere in the hot loop.
    const size_t rowbase = ((size_t)(valid ? b : 0) * 3 + (valid ? r : 0)) * ROWLEN;

    const float2* Yp = reinterpret_cast<const float2*>(labels + rowbase);
    const float2* Ap = reinterpret_cast<const float2*>(assign + rowbase);

    v8f acc = {};        // C accumulator: C[t_row][i_col] = dot(Y_t, D_i)
    float s1 = 0.0f;     // partial sum of clipped log(1-p) for this (b, i)

    // K = 100 = 25 WMMA steps of K=4. Lane (h,q) supplies A[M=q,K=2h+v] and
    // B[K=2h+v,N=q] in vector component v, per the 16x4 f32 VGPR layout.
    // Software-pipelined: prefetch next iteration's float2s before consuming.
    float2 y = Yp[h];                // it=0: kk = 2h -> index h
    float2 a = Ap[h];
    #pragma unroll
    for (int it = 0; it < 25; ++it) {
        float2 yn = {0.0f, 0.0f}, an = {0.0f, 0.0f};
        if (it < 24) {
            const int kk = (it + 1) * 4 + 2 * h;   // even -> 8B aligned
            yn = Yp[kk >> 1];
            an = Ap[kk >> 1];
        }

        const float lp0 = clog(a.x);
        const float l10 = clog(1.0f - a.x);
        const float lp1 = clog(a.y);
        const float l11 = clog(1.0f - a.y);

        s1 += l10 + l11;

        v2f av; av.x = y.x;        av.y = y.y;
        v2f bv; bv.x = lp0 - l10;  bv.y = lp1 - l11;

        // D = A(16x4, Y rows) * B(4x16, D cols) + C
        acc = __builtin_amdgcn_wmma_f32_16x16x4_f32(
                  false, av, false, bv, (short)0, acc, false, false);

        y = yn; a = an;
    }

    // Dump C tile (lane 0-15: M=v, lane 16-31: M=8+v, N=lane%16) and s1.
    #pragma unroll
    for (int v = 0; v < 8; ++v)
        ldsC[wv][(v + 8 * h) * 16 + q] = acc[v];
    ldsS[wv][lane] = s1;
    __syncthreads();

    // --- per-batch finalize: 6-perm min, argmin gather, category BCE ---
    if (lane < BATCHES_PER_WAVE) {
        const int bb = batch0 + lane;
        if (bb < NB) {
            const int base = 3 * lane;
            float cost[3][3];
            #pragma unroll
            for (int i = 0; i < 3; ++i) {
                const float s = ldsS[wv][base + i] + ldsS[wv][16 + base + i];
                #pragma unroll
                for (int t = 0; t < 3; ++t) {
                    const float dot = ldsC[wv][(base + t) * 16 + (base + i)];
                    cost[i][t] = -dot - s;      // sum of 100 BCE terms
                }
            }
            const int P[6][3] = {{0,1,2},{0,2,1},{1,0,2},{1,2,0},{2,0,1},{2,1,0}};
            float bestLoss = 3.402823e38f;
            int   best = 0;
            #pragma unroll
            for (int p = 0; p < 6; ++p) {
                const float l = (cost[P[p][0]][0] + cost[P[p][1]][1] +
                                 cost[P[p][2]][2]) * (1.0f / 300.0f);
                if (l < bestLoss) { bestLoss = l; best = p; }
            }
            float catsum = 0.0f;
            #pragma unroll
            for (int t = 0; t < 3; ++t) {
                const float pc = category[(size_t)bb * 3 + P[best][t]];
                const float yc = cat_labels[(size_t)bb * 3 + t];
                const float lp = clog(pc);
                const float l1 = clog(1.0f - pc);
                catsum += -(yc * lp + (1.0f - yc) * l1);
            }
            const float contrib = bestLoss * (1.0f / (float)NB)
                                + catsum  * (1.0f / (3.0f * (float)NB));
            atomicAdd(&blockSum, contrib);
        }
    }
    __syncthreads();
    if (tid == 0) atomicAdd(out, blockSum);
}

extern "C" void kernel_launch(void* const* d_in, const int* in_sizes, int n_in,
                              void* d_out, int out_size, void* d_ws, size_t ws_size,
                              hipStream_t stream) {
    (void)in_sizes; (void)n_in; (void)out_size; (void)d_ws; (void)ws_size;
    const float* assignments = (const float*)d_in[0];
    const float* category    = (const float*)d_in[1];
    const float* alabels     = (const float*)d_in[2];
    const float* clabels     = (const float*)d_in[3];
    float* out = (float*)d_out;

    hsl_zero_kernel<<<1, 1, 0, stream>>>(out);

    const int waves  = (NB + BATCHES_PER_WAVE - 1) / BATCHES_PER_WAVE;   // 13108
    const int blocks = (waves + WAVES_PER_BLOCK - 1) / WAVES_PER_BLOCK;  // 1639
    hsl_main_kernel<<<blocks, 256, 0, stream>>>(assignments, category,
                                                alabels, clabels, out);
}